// SRNN_50328426774674
// MI455X (gfx1250) — compile-verified
//
#include <hip/hip_runtime.h>
#include <math.h>

#define Tt 512
#define Bb 256
#define Dd 512
#define Hh 256
#define Oo 256
#define P1 112   // 100 padded to 112
#define P4 64    // 50  padded to 64
#define LDA 116  // LDS stride for 112-wide buffers (bank-conflict pad)
#define LDX 516  // LDS stride for 512-wide x tile (bank-conflict pad)

typedef __attribute__((ext_vector_type(2))) float v2f;
typedef __attribute__((ext_vector_type(8))) float v8f;

__device__ __forceinline__ v8f wmma4(v2f a, v2f b, v8f c) {
  // D = A(16x4 f32) x B(4x16 f32) + C(16x16 f32)
  return __builtin_amdgcn_wmma_f32_16x16x4_f32(false, a, false, b, (short)0, c,
                                               false, false);
}

// A row from LDS (per-lane row = M), B row from global weights [N][K] row-major.
__device__ __forceinline__ v8f gemm_tile(const float* __restrict__ arow, 
                                         const float* __restrict__ brow,
                                         int K, int half) {
  v8f c = {};
  for (int k = 0; k < K; k += 4) {
    int kk = k + half * 2;
    v2f a; a.x = arow[kk]; a.y = arow[kk + 1];
    v2f b; b.x = brow[kk]; b.y = brow[kk + 1];
    c = wmma4(a, b, c);
  }
  return c;
}

__device__ __forceinline__ float sigmoidf(float z) {
  return 1.0f / (1.0f + expf(-z));
}

// Zero-pad ragged weights to WMMA-friendly shapes.
__global__ void __launch_bounds__(256)
pad_weights(const float* __restrict__ wm1, const float* __restrict__ wm2,
            const float* __restrict__ wm3, const float* __restrict__ wm4,
            const float* __restrict__ wm5,
            float* __restrict__ w1p, float* __restrict__ w2p,
            float* __restrict__ w3p, float* __restrict__ w4p,
            float* __restrict__ w5p) {
  int idx = blockIdx.x * 256 + threadIdx.x;
  if (idx < P1 * Dd) {            // w1p: [112][512]
    int n = idx / Dd, k = idx % Dd;
    w1p[idx] = (n < 100) ? wm1[n * Dd + k] : 0.0f;
  }
  if (idx < P1 * P1) {            // w2p/w3p: [112][112]
    int n = idx / P1, k = idx % P1;
    bool ok = (n < 100) && (k < 100);
    w2p[idx] = ok ? wm2[n * 100 + k] : 0.0f;
    w3p[idx] = ok ? wm3[n * 100 + k] : 0.0f;
  }
  if (idx < P4 * P1) {            // w4p: [64][112]
    int n = idx / P1, k = idx % P1;
    w4p[idx] = ((n < 50) && (k < 100)) ? wm4[n * 100 + k] : 0.0f;
  }
  if (idx < Hh * P4) {            // w5p: [256][64]
    int n = idx / P4, k = idx % P4;
    w5p[idx] = (k < 50) ? wm5[n * 50 + k] : 0.0f;
  }
}

// One workgroup = one (t, 16-row batch tile). 8 waves, each owns 16x16 tiles.
__global__ void __launch_bounds__(256)
gate_kernel(const float* __restrict__ x,
            const float* __restrict__ w1p, const float* __restrict__ b1,
            const float* __restrict__ w2p, const float* __restrict__ b2,
            const float* __restrict__ w3p, const float* __restrict__ b3,
            const float* __restrict__ w4p, const float* __restrict__ b4,
            const float* __restrict__ w5p, const float* __restrict__ b5,
            const float* __restrict__ wx,  const float* __restrict__ bx,
            float* __restrict__ gate) {
  extern __shared__ float smem[];
  float* sX  = smem;                 // 16 * LDX
  float* sA  = sX + 16 * LDX;        // 16 * LDA  (a1 / a3)
  float* sB  = sA + 16 * LDA;        // 16 * LDA  (a2 / a4)
  float* sFX = sB + 16 * LDA;        // 16 * Hh   (fx)

  const int t   = blockIdx.x;
  const int m0  = blockIdx.y * 16;
  const int tid = threadIdx.x;
  const int wave = tid >> 5, lane = tid & 31;
  const int half = lane >> 4, l16 = lane & 15;

  // Stage x tile [16 x 512] into LDS (float4, padded row stride).
  const float4* __restrict__ xin =
      (const float4*)(x + ((size_t)t * Bb + m0) * Dd);
  for (int i = tid; i < 16 * (Dd / 4); i += 256) {
    int r = i >> 7, cc = i & 127;
    ((float4*)(sX + r * LDX))[cc] = xin[r * (Dd / 4) + cc];
  }
  __syncthreads();

  // ---- a1 = relu(x @ w1p^T + b1): N=112 (real 100), K=512
  for (int nt = wave; nt < 7; nt += 8) {
    v8f c = gemm_tile(sX + l16 * LDX, w1p + (nt * 16 + l16) * Dd, Dd, half);
    int col = nt * 16 + l16;
    float bias = (col < 100) ? b1[col] : 0.0f;
    for (int v = 0; v < 8; ++v) {
      int m = v + half * 8;
      sA[m * LDA + col] = (col < 100) ? fmaxf(c[v] + bias, 0.0f) : 0.0f;
    }
  }
  __syncthreads();

  // ---- a2 = relu(a1 @ w2p^T + b2): K=112
  for (int nt = wave; nt < 7; nt += 8) {
    v8f c = gemm_tile(sA + l16 * LDA, w2p + (nt * 16 + l16) * P1, P1, half);
    int col = nt * 16 + l16;
    float bias = (col < 100) ? b2[col] : 0.0f;
    for (int v = 0; v < 8; ++v) {
      int m = v + half * 8;
      sB[m * LDA + col] = (col < 100) ? fmaxf(c[v] + bias, 0.0f) : 0.0f;
    }
  }
  __syncthreads();

  // ---- a3 = relu(a2 @ w3p^T + b3): K=112
  for (int nt = wave; nt < 7; nt += 8) {
    v8f c = gemm_tile(sB + l16 * LDA, w3p + (nt * 16 + l16) * P1, P1, half);
    int col = nt * 16 + l16;
    float bias = (col < 100) ? b3[col] : 0.0f;
    for (int v = 0; v < 8; ++v) {
      int m = v + half * 8;
      sA[m * LDA + col] = (col < 100) ? fmaxf(c[v] + bias, 0.0f) : 0.0f;
    }
  }
  __syncthreads();

  // ---- a4 = relu(a3 @ w4p^T + b4): N=64 (real 50), K=112
  for (int nt = wave; nt < 4; nt += 8) {
    v8f c = gemm_tile(sA + l16 * LDA, w4p + (nt * 16 + l16) * P1, P1, half);
    int col = nt * 16 + l16;
    float bias = (col < 50) ? b4[col] : 0.0f;
    for (int v = 0; v < 8; ++v) {
      int m = v + half * 8;
      sB[m * LDA + col] = (col < 50) ? fmaxf(c[v] + bias, 0.0f) : 0.0f;
    }
  }
  __syncthreads();

  // ---- fx = sigmoid(a4 @ w5p^T + b5): N=256, K=64
  for (int nt = wave; nt < 16; nt += 8) {
    v8f c = gemm_tile(sB + l16 * LDA, w5p + (nt * 16 + l16) * P4, P4, half);
    int col = nt * 16 + l16;
    float bias = b5[col];
    for (int v = 0; v < 8; ++v) {
      int m = v + half * 8;
      sFX[m * Hh + col] = sigmoidf(c[v] + bias);
    }
  }
  __syncthreads();

  // ---- lx = x @ wx^T + bx; gate = fx * sigmoid(lx): N=256, K=512
  for (int nt = wave; nt < 16; nt += 8) {
    v8f c = gemm_tile(sX + l16 * LDX, wx + (nt * 16 + l16) * Dd, Dd, half);
    int col = nt * 16 + l16;
    float bias = bx[col];
    for (int v = 0; v < 8; ++v) {
      int m = v + half * 8;
      float g = sFX[m * Hh + col] * sigmoidf(c[v] + bias);
      gate[((size_t)t * Bb + m0 + m) * Hh + col] = g;
    }
  }
}

// Per-batch-row sequential recurrence: h = relu(roll(h,1) + gate[t]).
__global__ void __launch_bounds__(256)
recurrence_kernel(const float* __restrict__ gate, float* __restrict__ hidden_ws,
                  float* __restrict__ out_hidden) {
  __shared__ float sh[Hh];
  const int b = blockIdx.x, j = threadIdx.x;
  float h = 0.0f;
  const float* __restrict__ g = gate + (size_t)b * Hh + j;
  for (int t = 0; t < Tt; ++t) {
    sh[j] = h;
    __syncthreads();
    float prev = sh[(j + Hh - 1) & (Hh - 1)];   // roll(h, 1)[j] = h[j-1]
    h = fmaxf(prev + g[(size_t)t * Bb * Hh], 0.0f);
    __syncthreads();
  }
  hidden_ws[b * Hh + j] = h;
  out_hidden[b * Hh + j] = h;
}

// output = hidden @ wo^T + bo   (256x256x256, WMMA; 8 tiles per block)
__global__ void __launch_bounds__(256)
h2o_kernel(const float* __restrict__ hidden, const float* __restrict__ wo,
           const float* __restrict__ bo, float* __restrict__ out) {
  const int tid = threadIdx.x, wave = tid >> 5, lane = tid & 31;
  const int half = lane >> 4, l16 = lane & 15;
  const int tile = blockIdx.x * 8 + wave;  // 256 tiles total
  const int mt = tile & 15, nt = tile >> 4;
  v8f c = {};
  const float* __restrict__ arow = hidden + (mt * 16 + l16) * Hh;
  const float* __restrict__ brow = wo + (nt * 16 + l16) * Hh;
  for (int k = 0; k < Hh; k += 4) {
    int kk = k + half * 2;
    v2f a; a.x = arow[kk]; a.y = arow[kk + 1];
    v2f b; b.x = brow[kk]; b.y = brow[kk + 1];
    c = wmma4(a, b, c);
  }
  int col = nt * 16 + l16;
  float bias = bo[col];
  for (int v = 0; v < 8; ++v) {
    int m = mt * 16 + v + half * 8;
    out[m * Oo + col] = c[v] + bias;
  }
}

extern "C" void kernel_launch(void* const* d_in, const int* in_sizes, int n_in,
                              void* d_out, int out_size, void* d_ws,
                              size_t ws_size, hipStream_t stream) {
  (void)in_sizes; (void)n_in; (void)out_size; (void)ws_size;
  const float* x   = (const float*)d_in[0];
  const float* wm1 = (const float*)d_in[1];
  const float* bm1 = (const float*)d_in[2];
  const float* wm2 = (const float*)d_in[3];
  const float* bm2 = (const float*)d_in[4];
  const float* wm3 = (const float*)d_in[5];
  const float* bm3 = (const float*)d_in[6];
  const float* wm4 = (const float*)d_in[7];
  const float* bm4 = (const float*)d_in[8];
  const float* wm5 = (const float*)d_in[9];
  const float* bm5 = (const float*)d_in[10];
  const float* wx  = (const float*)d_in[11];
  const float* bx  = (const float*)d_in[12];
  const float* wo  = (const float*)d_in[13];
  const float* bo  = (const float*)d_in[14];

  float* out = (float*)d_out;  // [B*O] output, then [B*H] hidden

  float* ws = (float*)d_ws;
  size_t off = 0;
  float* gate   = ws + off; off += (size_t)Tt * Bb * Hh;  // 134 MB
  float* hidden = ws + off; off += (size_t)Bb * Hh;
  float* w1p    = ws + off; off += (size_t)P1 * Dd;
  float* w2p    = ws + off; off += (size_t)P1 * P1;
  float* w3p    = ws + off; off += (size_t)P1 * P1;
  float* w4p    = ws + off; off += (size_t)P4 * P1;
  float* w5p    = ws + off; off += (size_t)Hh * P4;

  pad_weights<<<(P1 * Dd + 255) / 256, 256, 0, stream>>>(
      wm1, wm2, wm3, wm4, wm5, w1p, w2p, w3p, w4p, w5p);

  size_t shmem = (size_t)(16 * LDX + 2 * 16 * LDA + 16 * Hh) * sizeof(float);
  gate_kernel<<<dim3(Tt, Bb / 16), 256, shmem, stream>>>(
      x, w1p, bm1, w2p, bm2, w3p, bm3, w4p, bm4, w5p, bm5, wx, bx, gate);

  recurrence_kernel<<<Bb, Hh, 0, stream>>>(gate, hidden, out + Bb * Oo);

  h2o_kernel<<<32, 256, 0, stream>>>(hidden, wo, bo, out);
}